// CumulativeProbingDense_29618094473736
// MI455X (gfx1250) — compile-verified
//
#include <hip/hip_runtime.h>
#include <math.h>

// Problem dims (fixed by the reference)
#define BB   16
#define LL   13
#define SS   1024
#define FF   768
#define HH   256
#define NLBL 7
#define MROWS (BB*SS)   // 16384

typedef __attribute__((ext_vector_type(2))) float v2f;
typedef __attribute__((ext_vector_type(8))) float v8f;

// ---------------------------------------------------------------------------
// Fused (optional layer-mix) GEMM:  C[row,col] = relu( A[row,:] . W[col,:] + bias[col] )
// A is MROWS x K (row-major). W is HH x K (row-major) -> B matrix is W^T.
// Block: 256 threads = 8 waves (2 M-waves x 4 N-waves). Block tile 64M x 256N.
// Wave tile 32M x 64N = 2x4 grid of 16x16 f32 WMMA accumulators.
// Double-buffered LDS + register prefetch: next K-slab's global loads are in
// flight while the current slab's 32 WMMAs execute.
// MIX=true: A is x (B,L,S,F); A-tile = gamma * sum_l softmax(mw)[l] * x[b,l,s,f].
// ---------------------------------------------------------------------------
template<int K, bool MIX>
__global__ __launch_bounds__(256) void gemm_bias_relu(
    const float* __restrict__ A, const float* __restrict__ W,
    const float* __restrict__ bias, float* __restrict__ C,
    const float* __restrict__ mw, const float* __restrict__ gamma,
    const int* __restrict__ layer)
{
    // Pad 18 -> every (row, even-k) pair is 8B aligned => ds_load_b64 fragments.
    __shared__ float As[2][64][18];
    // Bs[kpair][n][k&1]: B fragment for (kpair, n) is one contiguous aligned v2f.
    __shared__ float Bs[2][8][HH][2];

    const int tid  = threadIdx.x;
    const int lane = tid & 31;
    const int wv   = tid >> 5;       // wave id 0..7
    const int wm   = wv & 1;         // M-wave (0..1)
    const int wn   = wv >> 1;        // N-wave (0..3)
    const int lm   = lane & 15;      // lane's row/col within 16
    const int kh   = lane >> 4;      // K-half select (ISA 16x16x4 A/B layout)
    const int m0   = blockIdx.x * 64;

    // softmax(mixing_weights) * gamma (uniform; computed redundantly per thread)
    float w[LL];
    if (MIX) {
        float mx = -1e30f;
        #pragma unroll
        for (int l = 0; l < LL; ++l) mx = fmaxf(mx, mw[l]);
        float den = 0.f;
        #pragma unroll
        for (int l = 0; l < LL; ++l) { w[l] = __expf(mw[l] - mx); den += w[l]; }
        const float g  = gamma[0] / den;
        const int   nl = layer[0] + 1;
        #pragma unroll
        for (int l = 0; l < LL; ++l) w[l] = (l < nl) ? w[l] * g : 0.f;
    }

    v8f acc[2][4];
    #pragma unroll
    for (int mt = 0; mt < 2; ++mt)
        #pragma unroll
        for (int nt = 0; nt < 4; ++nt)
            #pragma unroll
            for (int i = 0; i < 8; ++i) acc[mt][nt][i] = 0.f;

    const int ar   = tid >> 2;       // A-tile row 0..63 loaded by this thread
    const int ac   = tid & 3;        // float4 column within 16
    const int arow = m0 + ar;        // global row = b*S + s
    const int bb   = arow >> 10;     // batch index (row / S)
    const int ss   = arow & (SS - 1);

    float4 pfA[MIX ? LL : 1];        // raw prefetch regs (A)
    float4 pfB[4];                   // raw prefetch regs (B)

    // Issue global loads for K-slab k0 into registers (no LDS yet).
    auto issue = [&](int k0) {
        if (MIX) {
            #pragma unroll
            for (int l = 0; l < LL; ++l)
                pfA[l] = *(const float4*)(
                    A + ((size_t)((bb * LL + l) * SS + ss)) * FF + k0 + ac * 4);
        } else {
            pfA[0] = *(const float4*)(A + (size_t)arow * K + k0 + ac * 4);
        }
        const float* wrow = W + (size_t)tid * K + k0;
        #pragma unroll
        for (int q = 0; q < 4; ++q)
            pfB[q] = *(const float4*)(wrow + q * 4);
    };

    // Reduce prefetch regs and commit to LDS buffer `buf`.
    auto commit = [&](int buf) {
        float4 a4;
        if (MIX) {
            a4.x = a4.y = a4.z = a4.w = 0.f;
            #pragma unroll
            for (int l = 0; l < LL; ++l) {
                a4.x += w[l] * pfA[l].x; a4.y += w[l] * pfA[l].y;
                a4.z += w[l] * pfA[l].z; a4.w += w[l] * pfA[l].w;
            }
        } else {
            a4 = pfA[0];
        }
        As[buf][ar][ac*4+0] = a4.x; As[buf][ar][ac*4+1] = a4.y;
        As[buf][ar][ac*4+2] = a4.z; As[buf][ar][ac*4+3] = a4.w;
        #pragma unroll
        for (int q = 0; q < 4; ++q) {           // k = 4q..4q+3
            Bs[buf][2*q    ][tid][0] = pfB[q].x;
            Bs[buf][2*q    ][tid][1] = pfB[q].y;
            Bs[buf][2*q + 1][tid][0] = pfB[q].z;
            Bs[buf][2*q + 1][tid][1] = pfB[q].w;
        }
    };

    // 4 WMMA k-steps over the 16-wide K slab in LDS buffer `buf`.
    auto compute = [&](int buf) {
        #pragma unroll
        for (int ks = 0; ks < 4; ++ks) {
            const int kp = ks * 2 + kh;         // k-pair; kk = 2*kp (lanes 16-31: +2)
            v2f afrag[2], bfrag[4];
            #pragma unroll
            for (int mt = 0; mt < 2; ++mt)
                afrag[mt] = *(const v2f*)&As[buf][wm*32 + mt*16 + lm][kp * 2];
            #pragma unroll
            for (int nt = 0; nt < 4; ++nt)
                bfrag[nt] = *(const v2f*)&Bs[buf][kp][wn*64 + nt*16 + lm][0];
            #pragma unroll
            for (int mt = 0; mt < 2; ++mt)
                #pragma unroll
                for (int nt = 0; nt < 4; ++nt)
                    acc[mt][nt] = __builtin_amdgcn_wmma_f32_16x16x4_f32(
                        false, afrag[mt], false, bfrag[nt],
                        (short)0, acc[mt][nt], false, false);
        }
    };

    // ---- software-pipelined main loop (one barrier per slab) ----
    issue(0);
    commit(0);
    __syncthreads();
    for (int k0 = 0; k0 < K; k0 += 16) {
        const int cur = (k0 >> 4) & 1;
        if (k0 + 16 < K) issue(k0 + 16);   // overlap with WMMAs below
        compute(cur);
        if (k0 + 16 < K) commit(cur ^ 1);
        __syncthreads();
    }

    // ---- epilogue: bias + ReLU + store ----
    // C/D layout: VGPR r holds row r (lanes 0-15) / row r+8 (lanes 16-31), col = lane%16
    #pragma unroll
    for (int nt = 0; nt < 4; ++nt) {
        const int col = wn*64 + nt*16 + lm;
        const float bv = bias[col];
        #pragma unroll
        for (int mt = 0; mt < 2; ++mt) {
            #pragma unroll
            for (int r = 0; r < 8; ++r) {
                const int row = m0 + wm*32 + mt*16 + ((lane < 16) ? r : r + 8);
                C[(size_t)row * HH + col] = fmaxf(acc[mt][nt][r] + bv, 0.f);
            }
        }
    }
}

// ---------------------------------------------------------------------------
// Masked mean-pool over S, then 7-way classifier head.
// One block per batch element b; thread h (0..255) owns channel h.
// ---------------------------------------------------------------------------
__global__ __launch_bounds__(256) void pool_logits(
    const float* __restrict__ h2, const int* __restrict__ lengths,
    const float* __restrict__ Wl, const float* __restrict__ bl,
    float* __restrict__ out)
{
    __shared__ float pooled[HH];
    const int b = blockIdx.x;
    const int h = threadIdx.x;
    const int len = lengths[b];

    float sum = 0.f;
    const float* p = h2 + (size_t)b * SS * HH + h;
    for (int s = 0; s < len; ++s) sum += p[(size_t)s * HH];
    pooled[h] = sum / (float)len;
    __syncthreads();

    if (h < NLBL) {
        float acc = bl[h];
        const float* wr = Wl + (size_t)h * HH;
        #pragma unroll 8
        for (int i = 0; i < HH; ++i) acc += pooled[i] * wr[i];
        out[b * NLBL + h] = acc;
    }
}

// ---------------------------------------------------------------------------
extern "C" void kernel_launch(void* const* d_in, const int* in_sizes, int n_in,
                              void* d_out, int out_size, void* d_ws, size_t ws_size,
                              hipStream_t stream) {
    const float* x       = (const float*)d_in[0];
    const int*   lengths = (const int*)  d_in[1];
    const int*   layer   = (const int*)  d_in[2];
    const float* gamma   = (const float*)d_in[3];
    const float* mw      = (const float*)d_in[4];
    const float* W1      = (const float*)d_in[5];
    const float* b1      = (const float*)d_in[6];
    const float* W2      = (const float*)d_in[7];
    const float* b2      = (const float*)d_in[8];
    const float* Wl      = (const float*)d_in[9];
    const float* bl      = (const float*)d_in[10];

    float* h1 = (float*)d_ws;                       // 16384 x 256
    float* h2 = h1 + (size_t)MROWS * HH;            // 16384 x 256
    float* out = (float*)d_out;                     // 16 x 7

    dim3 grid(MROWS / 64), block(256);
    // GEMM1: layer-mix fused into A-tile load (reads x exactly once)
    gemm_bias_relu<FF, true ><<<grid, block, 0, stream>>>(x,  W1, b1, h1, mw, gamma, layer);
    // GEMM2: h1 lives in L2
    gemm_bias_relu<HH, false><<<grid, block, 0, stream>>>(h1, W2, b2, h2, mw, gamma, layer);
    // Masked mean pool + classifier head
    pool_logits<<<dim3(BB), block, 0, stream>>>(h2, lengths, Wl, bl, out);
}